// BallisicProtectionGNN_89575837925845
// MI455X (gfx1250) — compile-verified
//
#include <hip/hip_runtime.h>
#include <hip/hip_bf16.h>
#include <math.h>

// ---------------------------------------------------------------------------
// BallisticProtectionGNN for MI455X (gfx1250, wave32, WMMA)
//  GCN(128->256) -> ReLU -> GCN(256->256) -> ReLU -> GAT(4 heads, mean) ->
//  ReLU -> mean-pool -> 4 linear heads.
// Dense GEMMs: V_WMMA_F32_16X16X4_F32 with the weight strip staged into LDS
// (async global->LDS when available), transposed so B fragments are single
// ds_load_b64 reads. Scatter phases are HBM/atomic bound: float4 gathers +
// f32 atomics.
// ---------------------------------------------------------------------------

typedef float v2f __attribute__((ext_vector_type(2)));
typedef float v8f __attribute__((ext_vector_type(8)));

#define HID    256
#define HEADS  4
#define GDIM   1024   // HEADS*HID
#define NGRAPH 32

#define GEMM_KC      64            // K-chunk staged in LDS
#define GEMM_PAD     4             // LDS row padding (floats) -> 8B align + no bank conflicts
#define GEMM_WAVES   5             // waves per block
#define GEMM_THREADS (GEMM_WAVES * 32)
#define GEMM_MROWS   (GEMM_WAVES * 16)   // 80 rows per block

#if __has_builtin(__builtin_amdgcn_global_load_async_to_lds_b32)
#define HAVE_ASYNC_LDS 1
typedef __attribute__((address_space(1))) int gas_int;   // global (AS1)
typedef __attribute__((address_space(3))) int las_int;   // LDS (AS3)
#endif

__device__ __forceinline__ void wait_async0() {
#if __has_builtin(__builtin_amdgcn_s_wait_asynccnt)
  __builtin_amdgcn_s_wait_asynccnt(0);
#else
  asm volatile("s_wait_asynccnt 0x0" ::: "memory");
#endif
}

__device__ __forceinline__ float atomic_max_f32(float* addr, float val) {
  if (val >= 0.0f)
    return __int_as_float(atomicMax((int*)addr, __float_as_int(val)));
  else
    return __uint_as_float(atomicMin((unsigned int*)addr, __float_as_uint(val)));
}

__global__ void k_fill(float* __restrict__ p, long n, float v) {
  long i = blockIdx.x * (long)blockDim.x + threadIdx.x;
  if (i < n) p[i] = v;
}

// ---------------------------------------------------------------------------
// C[M,N] = A[M,K] @ B[K,N] (f32, row-major).
// Block: 5 waves = 80(M) x 64(N); wave w owns rows [bm*80 + w*16, +16).
// B strip (K x 64) staged in LDS transposed: bt[n][k] so a fragment is one
// b64 read. A-frag: lane holds A[lane&15][2*(lane>>4)+v]; B-frag: lane holds
// B[k+2*(lane>>4)+v][lane&15]; C: vgpr r -> M=r+8*(lane>>4), N=lane&15.
// ---------------------------------------------------------------------------
__global__ __launch_bounds__(GEMM_THREADS)
void k_gemm_wmma(const float* __restrict__ A, const float* __restrict__ B,
                 float* __restrict__ C, int M, int N, int K) {
  __shared__ float bt[64][GEMM_KC + GEMM_PAD];

  const int tilesN = N >> 6;
  const int bm = blockIdx.x / tilesN;
  const int bn = blockIdx.x % tilesN;
  const int n0 = bn * 64;

  const int tid  = threadIdx.x;
  const int wv   = tid >> 5;
  const int lane = tid & 31;
  const int r16  = lane & 15;
  const int half = lane >> 4;

  int arow = bm * GEMM_MROWS + wv * 16 + r16;
  const bool mvalid = (bm * GEMM_MROWS + wv * 16) < M;   // wave-uniform
  if (arow >= M) arow = M - 1;                           // clamp: keep EXEC all-ones
  const float* Arow = A + (size_t)arow * K;

  v8f acc0 = {}, acc1 = {}, acc2 = {}, acc3 = {};

  for (int k0 = 0; k0 < K; k0 += GEMM_KC) {
    // ---- stage B[k0:k0+KC, n0:n0+64] into LDS, transposed ----
    for (int idx = tid; idx < GEMM_KC * 64; idx += GEMM_THREADS) {
      int j  = idx & 63;        // column within strip (coalesced global reads)
      int kk = idx >> 6;        // k within chunk
      size_t gidx = (size_t)(k0 + kk) * N + n0 + j;
#if HAVE_ASYNC_LDS
      __builtin_amdgcn_global_load_async_to_lds_b32(
          (gas_int*)(const_cast<float*>(B) + gidx),
          (las_int*)&bt[j][kk], 0, 0);
#else
      bt[j][kk] = B[gidx];
#endif
    }
#if HAVE_ASYNC_LDS
    wait_async0();
#endif
    __syncthreads();

    // ---- compute on the chunk ----
    for (int kk = 0; kk < GEMM_KC; kk += 4) {
      v2f a = *(const v2f*)(Arow + k0 + kk + 2 * half);
      int kb = kk + 2 * half;
      v2f b0 = *(const v2f*)&bt[r16 +  0][kb];
      v2f b1 = *(const v2f*)&bt[r16 + 16][kb];
      v2f b2 = *(const v2f*)&bt[r16 + 32][kb];
      v2f b3 = *(const v2f*)&bt[r16 + 48][kb];
      acc0 = __builtin_amdgcn_wmma_f32_16x16x4_f32(false, a, false, b0, (short)0, acc0, false, false);
      acc1 = __builtin_amdgcn_wmma_f32_16x16x4_f32(false, a, false, b1, (short)0, acc1, false, false);
      acc2 = __builtin_amdgcn_wmma_f32_16x16x4_f32(false, a, false, b2, (short)0, acc2, false, false);
      acc3 = __builtin_amdgcn_wmma_f32_16x16x4_f32(false, a, false, b3, (short)0, acc3, false, false);
    }
    __syncthreads();
  }

  if (!mvalid) return;
  for (int r = 0; r < 8; ++r) {
    int m = bm * GEMM_MROWS + wv * 16 + r + 8 * half;
    if (m >= M) continue;
    float* Crow = C + (size_t)m * N + n0 + r16;
    Crow[0]  = acc0[r];
    Crow[16] = acc1[r];
    Crow[32] = acc2[r];
    Crow[48] = acc3[r];
  }
}

// ------------------------- GCN pieces --------------------------------------
__global__ void k_deg(const int* __restrict__ ei, float* __restrict__ deg, int E, int Nn) {
  int e = blockIdx.x * blockDim.x + threadIdx.x;
  if (e >= E + Nn) return;
  int dst = (e < E) ? ei[E + e] : (e - E);   // self loops appended
  atomicAdd(deg + dst, 1.0f);
}

__global__ void k_dinv(const float* __restrict__ deg, float* __restrict__ dinv, int Nn) {
  int i = blockIdx.x * blockDim.x + threadIdx.x;
  if (i < Nn) dinv[i] = rsqrtf(fmaxf(deg[i], 1e-12f));
}

// out[dst,:] += h[src,:] * dinv[src]*dinv[dst]; 64 float4 chunks per edge (C=256)
__global__ void k_gcn_scatter(const float* __restrict__ h, const float* __restrict__ dinv,
                              const int* __restrict__ ei, float* __restrict__ out,
                              int E, int Nn) {
  long gid = blockIdx.x * (long)blockDim.x + threadIdx.x;
  long tot = (long)(E + Nn) * 64;
  if (gid >= tot) return;
  int e = (int)(gid >> 6);
  int c = (int)(gid & 63) << 2;
  int src, dst;
  if (e < E) { src = ei[e]; dst = ei[E + e]; } else { src = dst = e - E; }
  float coef = dinv[src] * dinv[dst];
  float4 hv = *(const float4*)(h + (size_t)src * HID + c);
  float* o = out + (size_t)dst * HID + c;
  atomicAdd(o + 0, hv.x * coef);
  atomicAdd(o + 1, hv.y * coef);
  atomicAdd(o + 2, hv.z * coef);
  atomicAdd(o + 3, hv.w * coef);
}

__global__ void k_bias_relu(float* __restrict__ p, const float* __restrict__ b, long tot, int C) {
  long i = blockIdx.x * (long)blockDim.x + threadIdx.x;
  if (i >= tot) return;
  float v = p[i] + b[i % C];
  p[i] = v > 0.0f ? v : 0.0f;
}

// ------------------------- GAT pieces --------------------------------------
__global__ void k_gat_scores(const float* __restrict__ hg, const float* __restrict__ att_s,
                             const float* __restrict__ att_d,
                             float* __restrict__ as_, float* __restrict__ ad_, int Nn) {
  int i = blockIdx.x * blockDim.x + threadIdx.x;       // over Nn*HEADS
  if (i >= Nn * HEADS) return;
  int n = i / HEADS, h = i % HEADS;
  const float* row = hg + (size_t)n * GDIM + h * HID;
  const float* ws  = att_s + h * HID;
  const float* wd  = att_d + h * HID;
  float ss = 0.0f, sd = 0.0f;
  for (int c = 0; c < HID; ++c) { float v = row[c]; ss += v * ws[c]; sd += v * wd[c]; }
  as_[i] = ss; ad_[i] = sd;
}

__device__ __forceinline__ void edge_src_dst(const int* ei, int e, int E, int& src, int& dst) {
  if (e < E) { src = ei[e]; dst = ei[E + e]; } else { src = dst = e - E; }
}

__global__ void k_edge_max(const float* __restrict__ as_, const float* __restrict__ ad_,
                           const int* __restrict__ ei, float* __restrict__ m, int E, int Nn) {
  long gid = blockIdx.x * (long)blockDim.x + threadIdx.x;
  long tot = (long)(E + Nn) * HEADS;
  if (gid >= tot) return;
  int e = (int)(gid / HEADS), h = (int)(gid % HEADS);
  int src, dst; edge_src_dst(ei, e, E, src, dst);
  float s = as_[src * HEADS + h] + ad_[dst * HEADS + h];
  s = s > 0.0f ? s : 0.2f * s;                         // leaky_relu(0.2)
  atomic_max_f32(m + dst * HEADS + h, s);
}

__global__ void k_edge_exp(const float* __restrict__ as_, const float* __restrict__ ad_,
                           const int* __restrict__ ei, const float* __restrict__ m,
                           float* __restrict__ z, float* __restrict__ eb, int E, int Nn) {
  long gid = blockIdx.x * (long)blockDim.x + threadIdx.x;
  long tot = (long)(E + Nn) * HEADS;
  if (gid >= tot) return;
  int e = (int)(gid / HEADS), h = (int)(gid % HEADS);
  int src, dst; edge_src_dst(ei, e, E, src, dst);
  float s = as_[src * HEADS + h] + ad_[dst * HEADS + h];
  s = s > 0.0f ? s : 0.2f * s;
  float ex = __expf(s - m[dst * HEADS + h]);
  eb[gid] = ex;
  atomicAdd(z + dst * HEADS + h, ex);
}

// out[dst, h*256+c..] += hg[src, h*256+c..] * alpha(e,h)
__global__ void k_gat_scatter(const float* __restrict__ hg, const float* __restrict__ eb,
                              const float* __restrict__ z, const int* __restrict__ ei,
                              float* __restrict__ out, int E, int Nn) {
  long gid = blockIdx.x * (long)blockDim.x + threadIdx.x;
  long tot = (long)(E + Nn) * HEADS * 64;
  if (gid >= tot) return;
  int  chunk = (int)(gid & 63);
  long eh    = gid >> 6;                     // e*HEADS + h
  int  h = (int)(eh % HEADS);
  int  e = (int)(eh / HEADS);
  int src, dst; edge_src_dst(ei, e, E, src, dst);
  float alpha = eb[eh] / (z[dst * HEADS + h] + 1e-16f);
  int c = h * HID + chunk * 4;
  float4 v = *(const float4*)(hg + (size_t)src * GDIM + c);
  float* o = out + (size_t)dst * GDIM + c;
  atomicAdd(o + 0, v.x * alpha);
  atomicAdd(o + 1, v.y * alpha);
  atomicAdd(o + 2, v.z * alpha);
  atomicAdd(o + 3, v.w * alpha);
}

__global__ void k_head_mean_relu(const float* __restrict__ g, const float* __restrict__ bg,
                                 float* __restrict__ h3, int Nn) {
  long i = blockIdx.x * (long)blockDim.x + threadIdx.x;
  long tot = (long)Nn * HID;
  if (i >= tot) return;
  int n = (int)(i / HID), c = (int)(i % HID);
  const float* r = g + (size_t)n * GDIM + c;
  float v = 0.25f * (r[0] + r[HID] + r[2 * HID] + r[3 * HID]) + bg[c];
  h3[i] = v > 0.0f ? v : 0.0f;
}

// ------------------------- pooling + heads ---------------------------------
__global__ void k_pool(const float* __restrict__ h3, const int* __restrict__ batch,
                       float* __restrict__ pooled, float* __restrict__ cnt, int Nn) {
  long i = blockIdx.x * (long)blockDim.x + threadIdx.x;
  long tot = (long)Nn * HID;
  if (i >= tot) return;
  int n = (int)(i / HID), c = (int)(i % HID);
  int b = batch[n];
  atomicAdd(pooled + (size_t)b * HID + c, h3[i]);
  if (c == 0) atomicAdd(cnt + b, 1.0f);
}

__global__ void k_heads(const float* __restrict__ pooled, const float* __restrict__ cnt,
                        const float* __restrict__ Wh, const float* __restrict__ bh,
                        float* __restrict__ out) {
  int i = threadIdx.x;                         // 0..127 -> [32 graphs x 4 heads]
  if (i >= NGRAPH * 4) return;
  int g = i >> 2, j = i & 3;
  float inv = 1.0f / fmaxf(cnt[g], 1.0f);
  float s = 0.0f;
  const float* pr = pooled + (size_t)g * HID;
  for (int c = 0; c < HID; ++c) s += pr[c] * inv * Wh[c * 4 + j];
  out[i] = s + bh[j];
}

// ---------------------------------------------------------------------------
extern "C" void kernel_launch(void* const* d_in, const int* in_sizes, int n_in,
                              void* d_out, int out_size, void* d_ws, size_t ws_size,
                              hipStream_t stream) {
  const float* x     = (const float*)d_in[0];
  const int*   ei    = (const int*)d_in[1];
  const int*   batch = (const int*)d_in[2];
  const float* W1    = (const float*)d_in[3];
  const float* b1    = (const float*)d_in[4];
  const float* W2    = (const float*)d_in[5];
  const float* b2    = (const float*)d_in[6];
  const float* Wg    = (const float*)d_in[7];
  const float* att_s = (const float*)d_in[8];
  const float* att_d = (const float*)d_in[9];
  const float* bg    = (const float*)d_in[10];
  const float* Wh    = (const float*)d_in[11];
  const float* bh    = (const float*)d_in[12];
  float* out = (float*)d_out;

  const int Nn = in_sizes[2];            // 10000 nodes
  const int E  = in_sizes[1] / 2;        // 160000 edges
  const int IN = in_sizes[0] / Nn;       // 128

  // ---- workspace layout (floats), ~116 MB total ----
  float* ws   = (float*)d_ws;
  float* bufA = ws;                                  // Nn*HID  (h1pre / h2pre)
  float* bufB = bufA + (size_t)Nn * HID;             // Nn*HID  (h1 / h2)
  float* hg   = bufB + (size_t)Nn * HID;             // Nn*GDIM
  float* gat  = hg   + (size_t)Nn * GDIM;            // Nn*GDIM
  float* h3   = gat  + (size_t)Nn * GDIM;            // Nn*HID
  float* deg  = h3   + (size_t)Nn * HID;             // Nn
  float* dinv = deg  + Nn;                           // Nn
  float* as_  = dinv + Nn;                           // Nn*HEADS
  float* ad_  = as_  + (size_t)Nn * HEADS;           // Nn*HEADS
  float* mb   = ad_  + (size_t)Nn * HEADS;           // Nn*HEADS
  float* zb   = mb   + (size_t)Nn * HEADS;           // Nn*HEADS
  float* ebuf = zb   + (size_t)Nn * HEADS;           // (E+Nn)*HEADS
  float* pooled = ebuf + (size_t)(E + Nn) * HEADS;   // 32*HID
  float* cnt  = pooled + NGRAPH * HID;               // 32

  auto nb = [](long n, int t) { return (int)((n + t - 1) / t); };
  const int T = 256;
  const int gemmBlocksM = (Nn + GEMM_MROWS - 1) / GEMM_MROWS;   // 125

  // ---- degree / normalization (shared by both GCN layers) ----
  k_fill<<<nb(Nn, T), T, 0, stream>>>(deg, Nn, 0.0f);
  k_deg<<<nb(E + Nn, T), T, 0, stream>>>(ei, deg, E, Nn);
  k_dinv<<<nb(Nn, T), T, 0, stream>>>(deg, dinv, Nn);

  // ---- GCN layer 1: h1 = relu(scatter(x@W1) + b1) ----
  k_gemm_wmma<<<gemmBlocksM * (HID / 64), GEMM_THREADS, 0, stream>>>(x, W1, bufA, Nn, HID, IN);
  k_fill<<<nb((long)Nn * HID, T), T, 0, stream>>>(bufB, (long)Nn * HID, 0.0f);
  k_gcn_scatter<<<nb((long)(E + Nn) * 64, T), T, 0, stream>>>(bufA, dinv, ei, bufB, E, Nn);
  k_bias_relu<<<nb((long)Nn * HID, T), T, 0, stream>>>(bufB, b1, (long)Nn * HID, HID);

  // ---- GCN layer 2: h2 = relu(scatter(h1@W2) + b2) ----
  k_gemm_wmma<<<gemmBlocksM * (HID / 64), GEMM_THREADS, 0, stream>>>(bufB, W2, bufA, Nn, HID, HID);
  k_fill<<<nb((long)Nn * HID, T), T, 0, stream>>>(bufB, (long)Nn * HID, 0.0f);
  k_gcn_scatter<<<nb((long)(E + Nn) * 64, T), T, 0, stream>>>(bufA, dinv, ei, bufB, E, Nn);
  k_bias_relu<<<nb((long)Nn * HID, T), T, 0, stream>>>(bufB, b2, (long)Nn * HID, HID);

  // ---- GAT: hg = h2@Wg ; edge softmax ; weighted scatter ; head mean ----
  k_gemm_wmma<<<gemmBlocksM * (GDIM / 64), GEMM_THREADS, 0, stream>>>(bufB, Wg, hg, Nn, GDIM, HID);
  k_gat_scores<<<nb((long)Nn * HEADS, T), T, 0, stream>>>(hg, att_s, att_d, as_, ad_, Nn);
  k_fill<<<nb((long)Nn * HEADS, T), T, 0, stream>>>(mb, (long)Nn * HEADS, -INFINITY);
  k_fill<<<nb((long)Nn * HEADS, T), T, 0, stream>>>(zb, (long)Nn * HEADS, 0.0f);
  k_edge_max<<<nb((long)(E + Nn) * HEADS, T), T, 0, stream>>>(as_, ad_, ei, mb, E, Nn);
  k_edge_exp<<<nb((long)(E + Nn) * HEADS, T), T, 0, stream>>>(as_, ad_, ei, mb, zb, ebuf, E, Nn);
  k_fill<<<nb((long)Nn * GDIM, T), T, 0, stream>>>(gat, (long)Nn * GDIM, 0.0f);
  k_gat_scatter<<<nb((long)(E + Nn) * HEADS * 64, T), T, 0, stream>>>(hg, ebuf, zb, ei, gat, E, Nn);
  k_head_mean_relu<<<nb((long)Nn * HID, T), T, 0, stream>>>(gat, bg, h3, Nn);

  // ---- mean pool + regression heads ----
  k_fill<<<nb(NGRAPH * HID, T), T, 0, stream>>>(pooled, NGRAPH * HID, 0.0f);
  k_fill<<<1, 32, 0, stream>>>(cnt, NGRAPH, 0.0f);
  k_pool<<<nb((long)Nn * HID, T), T, 0, stream>>>(h3, batch, pooled, cnt, Nn);
  k_heads<<<1, 128, 0, stream>>>(pooled, cnt, Wh, bh, out);
}